// LambdaRankLoss_40355512713943
// MI455X (gfx1250) — compile-verified
//
#include <hip/hip_runtime.h>
#include <hip/hip_bf16.h>

// LambdaRank pairwise loss, B=256 rows x N=512 items.
// Kernel 1: one block per row.  softplus(s_j - s_i) = log(1 + e^{s_j-m} * e^{m-s_i})
//           with m = row max, so the 512x512 inner loop needs only ONE
//           transcendental (v_log_f32) + two FMAs per pair.  Per-block
//           reduction of the 256 f32 partials via V_WMMA_F32_16X16X4_F32
//           (ones-matrix x partials, accumulated over 4 chunks).
// Kernel 2: deterministic tree-combine of per-row partials -> scalar output.

typedef __attribute__((ext_vector_type(2))) float v2f;
typedef __attribute__((ext_vector_type(8))) float v8f;

#define NITEMS 512
#define BLOCK  256
#define IPT    (NITEMS / BLOCK)   // i-rows per thread (= 2)

__global__ __launch_bounds__(BLOCK) void lambdarank_row_kernel(
    const float* __restrict__ scores,
    const int* __restrict__ relevances,
    float* __restrict__ wsLoss,
    unsigned int* __restrict__ wsCnt)
{
    __shared__ float2 sEM[NITEMS];     // {exp(s_j - m), negMask}
    __shared__ float  red[BLOCK];      // row-max scratch, then partial losses
    __shared__ int    sPosCnt;

    const int t = threadIdx.x;
    const int b = blockIdx.x;
    const float* srow = scores + (size_t)b * NITEMS;
    const int*   rrow = relevances + (size_t)b * NITEMS;

    if (t == 0) sPosCnt = 0;
    __syncthreads();

    float sreg[IPT];
    int   rreg[IPT];
    int myPos = 0;
    #pragma unroll
    for (int u = 0; u < IPT; ++u) {
        int i = t + u * BLOCK;
        sreg[u] = srow[i];
        rreg[u] = rrow[i];
        myPos += (rreg[u] == 1) ? 1 : 0;
    }
    atomicAdd(&sPosCnt, myPos);

    // Block-wide row max (deterministic tree).
    red[t] = fmaxf(sreg[0], sreg[1]);
    __syncthreads();
    #pragma unroll
    for (int s = BLOCK / 2; s > 0; s >>= 1) {
        if (t < s) red[t] = fmaxf(red[t], red[t + s]);
        __syncthreads();
    }
    const float m = red[0];
    __syncthreads();   // red[] will be reused for loss partials

    // Stage {e_j, negmask} into LDS.  e_j = exp(s_j - m) <= 1, no overflow.
    #pragma unroll
    for (int u = 0; u < IPT; ++u) {
        int i = t + u * BLOCK;
        float2 v;
        v.x = __expf(sreg[u] - m);
        v.y = (rreg[u] == 0) ? 1.0f : 0.0f;
        sEM[i] = v;
    }
    __syncthreads();

    // Each thread owns i = t and i = t + 256; for each positive i:
    //   pair term = log(1 + e_j * p_i),  p_i = exp(m - s_i).
    // Inner loop: 1x ds_load_b64 + v_fma + v_log_f32 + v_fma.
    float partial = 0.0f;
    #pragma unroll
    for (int u = 0; u < IPT; ++u) {
        if (rreg[u] == 1) {
            const float pi = __expf(m - sreg[u]);
            float acc = 0.0f;
            #pragma unroll 8
            for (int j = 0; j < NITEMS; ++j) {
                float2 v = sEM[j];              // uniform index -> LDS broadcast
                float l = __logf(__builtin_fmaf(v.x, pi, 1.0f));
                acc = __builtin_fmaf(v.y, l, acc);
            }
            partial += acc;
        }
    }

    red[t] = partial;
    __syncthreads();

    // WMMA reduction of 256 f32 partials by wave 0 (EXEC all ones for the
    // 32-lane wave).  D = sum_{chunks} ones(16x4) x B(4x16): D[0][n] = column
    // sums of the 16x16 partial tile; 4 shfl_xor steps finish lanes 0..15.
    if (t < 32) {
        const int n     = t & 15;   // B column
        const int khalf = t >> 4;   // lanes 16-31 carry K=2,3
        v2f a; a.x = 1.0f; a.y = 1.0f;
        v8f c = {};
        #pragma unroll
        for (int ch = 0; ch < 4; ++ch) {
            int k0 = ch * 4 + khalf * 2;
            v2f bv;
            bv.x = red[(k0 + 0) * 16 + n];
            bv.y = red[(k0 + 1) * 16 + n];
            c = __builtin_amdgcn_wmma_f32_16x16x4_f32(
                    /*neg_a=*/false, a, /*neg_b=*/false, bv,
                    /*c_mod=*/(short)0, c, /*reuse_a=*/false, /*reuse_b=*/false);
        }
        float v = c[0];                      // row M=0: colsum[n] in lanes 0..15
        v += __shfl_xor(v, 8, 32);
        v += __shfl_xor(v, 4, 32);
        v += __shfl_xor(v, 2, 32);
        v += __shfl_xor(v, 1, 32);
        if (t == 0) {
            int nPos = sPosCnt;
            wsLoss[b] = v;
            wsCnt[b]  = (unsigned int)nPos * (unsigned int)(NITEMS - nPos);
        }
    }
}

__global__ __launch_bounds__(BLOCK) void lambdarank_finalize_kernel(
    const float* __restrict__ wsLoss,
    const unsigned int* __restrict__ wsCnt,
    float* __restrict__ out,
    int nrows)
{
    __shared__ float        L[BLOCK];
    __shared__ unsigned int C[BLOCK];
    const int t = threadIdx.x;
    L[t] = (t < nrows) ? wsLoss[t] : 0.0f;
    C[t] = (t < nrows) ? wsCnt[t] : 0u;
    __syncthreads();
    #pragma unroll
    for (int s = BLOCK / 2; s > 0; s >>= 1) {
        if (t < s) { L[t] += L[t + s]; C[t] += C[t + s]; }
        __syncthreads();
    }
    if (t == 0) {
        float total = L[0];
        unsigned int cnt = C[0];
        out[0] = (cnt > 0u) ? (total / (float)cnt) : total;
    }
}

extern "C" void kernel_launch(void* const* d_in, const int* in_sizes, int n_in,
                              void* d_out, int out_size, void* d_ws, size_t ws_size,
                              hipStream_t stream) {
    const float* scores     = (const float*)d_in[0];
    const int*   relevances = (const int*)d_in[1];
    float*       out        = (float*)d_out;

    const int B = in_sizes[0] / NITEMS;   // 256 for the reference shapes

    float*        wsLoss = (float*)d_ws;
    unsigned int* wsCnt  = (unsigned int*)(wsLoss + ((B + 63) & ~63));

    lambdarank_row_kernel<<<B, BLOCK, 0, stream>>>(scores, relevances, wsLoss, wsCnt);
    lambdarank_finalize_kernel<<<1, BLOCK, 0, stream>>>(wsLoss, wsCnt, out, B);
}